// NodeToSeq_v2_71330816852465
// MI455X (gfx1250) — compile-verified
//
#include <hip/hip_runtime.h>
#include <hip/hip_bf16.h>
#include <stdint.h>

#define D_DIM 512
#define B_DIM 64
#define L_DIM 2048
#define QSPLIT 8     // row-stream splits per bucket (gather kernel gridDim.x)
#define BCHUNK 128   // sequence positions per broadcast workgroup
#define TROWS  16    // rows per TDM store tile (BCHUNK / 8 waves)

typedef unsigned int v4u __attribute__((ext_vector_type(4)));
typedef int          v8i __attribute__((ext_vector_type(8)));
typedef int          v4i __attribute__((ext_vector_type(4)));

__device__ __forceinline__ void f4add(float4& a, const float4 b) {
  a.x += b.x; a.y += b.y; a.z += b.z; a.w += b.w;
}

// ---------------------------------------------------------------------------
// Stage 1: gather-accumulate. WG (q, b) scans token_position[jbeg..jend) and
// register-accumulates matching g2 rows (g2[j] == gcn[j>>1]) for bucket b.
// ---------------------------------------------------------------------------
__global__ __launch_bounds__(256) void gather_accum_kernel(
    const float* __restrict__ g, const long long* __restrict__ tp,
    int twoN, float* __restrict__ xacc) {
  const int b     = blockIdx.y;
  const int chunk = (twoN + QSPLIT - 1) / QSPLIT;
  const int jbeg  = blockIdx.x * chunk;
  const int jend  = min(jbeg + chunk, twoN);
  const int wave  = threadIdx.x >> 5;
  const int lane  = threadIdx.x & 31;

  float4 a0 = {0.f,0.f,0.f,0.f}, a1 = {0.f,0.f,0.f,0.f};
  float4 a2 = {0.f,0.f,0.f,0.f}, a3 = {0.f,0.f,0.f,0.f};

  for (int base = jbeg + wave * 32; base < jend; base += 8 * 32) {
    const int j = base + lane;
    const bool m = (j < jend) && (tp[j] == (long long)b);
    unsigned mask = __builtin_amdgcn_ballot_w32(m);
    while (mask) {                       // wave-uniform loop over matches
      const int bit = __builtin_ctz(mask);
      mask &= mask - 1;
      const float4* row =
          (const float4*)(g + (size_t)((base + bit) >> 1) * D_DIM);
      f4add(a0, row[lane]);              // 4 x global_load_b128, 512B coalesced
      f4add(a1, row[lane + 32]);
      f4add(a2, row[lane + 64]);
      f4add(a3, row[lane + 96]);
    }
  }

  // Per-wave partials -> LDS (disjoint, no atomics), then 512 global f32 adds.
  __shared__ float4 xs[8][128];          // 16 KB
  xs[wave][lane      ] = a0;
  xs[wave][lane +  32] = a1;
  xs[wave][lane +  64] = a2;
  xs[wave][lane +  96] = a3;
  __syncthreads();
  const float* xsf = (const float*)xs;
  for (int i = threadIdx.x; i < D_DIM; i += 256) {
    float s = 0.f;
#pragma unroll
    for (int w = 0; w < 8; ++w) s += xsf[w * D_DIM + i];
    unsafeAtomicAdd(&xacc[(size_t)b * D_DIM + i], s);   // global_atomic_add_f32
  }
}

// ---------------------------------------------------------------------------
// Stage 2: broadcast x[b] to L sequence slots.
//   1) async-load x[b] (2 KB) into LDS row 0            (ASYNCcnt path)
//   2) replicate to a 16x512 LDS tile (32 KB)
//   3) each wave fires ONE tensor_store_from_lds (TDM)  (TENSORcnt path)
//      writing the 32 KB tile to its 16-row output slot.
// ---------------------------------------------------------------------------
__global__ __launch_bounds__(256) void broadcast_kernel(
    const float* __restrict__ x, float* __restrict__ out) {
  const int b  = blockIdx.y;
  const int l0 = blockIdx.x * BCHUNK;
  const int t  = threadIdx.x;
  __shared__ float tile[TROWS * D_DIM];  // 32 KB

  const int c16 = t & 127;               // which 16-byte chunk of a row
  const unsigned ldsRow0 = (unsigned)(uintptr_t)(&tile[c16 * 4]);

  if (t < 128) {
    const unsigned long long src =
        (unsigned long long)(uintptr_t)(x + (size_t)b * D_DIM + (size_t)c16 * 4);
    asm volatile("global_load_async_to_lds_b128 %0, %1, off"
                 :: "v"(ldsRow0), "v"(src) : "memory");
  }
  asm volatile("s_wait_asynccnt 0x0" ::: "memory");
  __syncthreads();

  // replicate row 0 into rows 1..15 (16 B stores, bank-conflict free)
  const float4 v = ((const float4*)tile)[c16];
  for (int r = 1 + (t >> 7); r < TROWS; r += 2)
    ((float4*)tile)[r * 128 + c16] = v;
  __syncthreads();

  // One TDM descriptor per wave: 2D tile 512x16 f32, LDS -> global.
  const int w = t >> 5;                  // wave id (wave-uniform)
  const unsigned long long ga = (unsigned long long)(uintptr_t)
      (out + (((size_t)b * L_DIM) + (size_t)(l0 + w * TROWS)) * D_DIM);
  const unsigned ldsTile = (unsigned)(uintptr_t)tile;

  // D# group 0: count=1 | lds_addr | global_addr(57b) | type=2
  const v4u g0 = { 1u,
                   ldsTile,
                   (unsigned)(ga & 0xffffffffu),
                   (unsigned)((ga >> 32) & 0x1ffffffu) | 0x80000000u };
  // D# group 1: data_size=2 (4B); dims/tile 512x16; dim0 stride 512
  const v8i g1 = { (int)(2u << 16),     // [17:16] data_size=4B, no iterate/pad
                   (int)(512u << 16),   // [79:48]  tensor_dim0 = 512
                   (int)(16u  << 16),   // [111:80] tensor_dim1 = 16
                   (int)(512u << 16),   // [127:112] tile_dim0 = 512
                   16,                  // [143:128] tile_dim1 = 16 (tile_dim2=0)
                   512,                 // [207:160] tensor_dim0_stride = 512
                   0, 0 };              // tensor_dim1_stride = 0 (unused, 2D)
  const v4i g2 = { 0, 0, 0, 0 };        // 2D: higher dims unused
  const v4i g3 = { 0, 0, 0, 0 };
  const v8i g4 = { 0, 0, 0, 0, 0, 0, 0, 0 };  // unused VADDR4 slot (NULL group)
  __builtin_amdgcn_tensor_store_from_lds(g0, g1, g2, g3, g4, 0);
  __builtin_amdgcn_s_wait_tensorcnt(0);
}

// ---------------------------------------------------------------------------
extern "C" void kernel_launch(void* const* d_in, const int* in_sizes, int n_in,
                              void* d_out, int out_size, void* d_ws, size_t ws_size,
                              hipStream_t stream) {
  const float*     g    = (const float*)d_in[0];      // [N, 512] f32
  const long long* tp   = (const long long*)d_in[1];  // [2N] i64, values in [0,64)
  const int        twoN = in_sizes[1];
  float*           xacc = (float*)d_ws;               // [64, 512] f32 accumulator
  float*           out  = (float*)d_out;              // [64, 2048, 512] f32

  (void)hipMemsetAsync(xacc, 0, (size_t)B_DIM * D_DIM * sizeof(float), stream);

  gather_accum_kernel<<<dim3(QSPLIT, B_DIM), 256, 0, stream>>>(g, tp, twoN, xacc);

  broadcast_kernel<<<dim3(L_DIM / BCHUNK, B_DIM), 256, 0, stream>>>(xacc, out);
}